// SGC_LL_89489938579862
// MI455X (gfx1250) — compile-verified
//
#include <hip/hip_runtime.h>

typedef __attribute__((ext_vector_type(2))) float v2f;
typedef __attribute__((ext_vector_type(8))) float v8f;

#define BMOL 64
#define MM   512
#define FINN 128
#define OUTN 128
#define SLOPE 0.01f

__device__ __forceinline__ float leaky(float v) { return v >= 0.0f ? v : SLOPE * v; }

// one k-step (depth 4) of fp32 WMMA: c += a(16x4) * b(4x16)
#define WMMA_F32(cacc, av, bv) \
    cacc = __builtin_amdgcn_wmma_f32_16x16x4_f32(false, av, false, bv, (short)0, cacc, false, false)

// ============ Kernel 1: xw = x @ M_L, 2x2 blocked (32x32 per wave) ============
// blocks of rows: 512/32 = 16 ; cols: 128/32 = 4 ; waves = B*16*4
__global__ void k_xw(const float* __restrict__ x, const float* __restrict__ ML,
                     float* __restrict__ xw) {
    const int lane = threadIdx.x & 31;
    const int wave = threadIdx.x >> 5;
    const int t    = blockIdx.x * 4 + wave;
    const int b    = t >> 6;            // / (16*4)
    const int ib   = (t >> 2) & 15;
    const int jb   = t & 3;
    const int half = lane >> 4;
    const int r    = lane & 15;

    const float* A = x + (size_t)b * MM * FINN;
    const int arow0 = ib * 32 + r, arow1 = arow0 + 16;
    const int bcol0 = jb * 32 + r, bcol1 = bcol0 + 16;
    v8f c00 = {}, c01 = {}, c10 = {}, c11 = {};
    for (int k = 0; k < FINN; k += 4) {
        const int kk = k + 2 * half;
        v2f a0, a1, b0, b1;
        a0[0] = A[arow0 * FINN + kk];     a0[1] = A[arow0 * FINN + kk + 1];
        a1[0] = A[arow1 * FINN + kk];     a1[1] = A[arow1 * FINN + kk + 1];
        b0[0] = ML[kk * FINN + bcol0];    b0[1] = ML[(kk + 1) * FINN + bcol0];
        b1[0] = ML[kk * FINN + bcol1];    b1[1] = ML[(kk + 1) * FINN + bcol1];
        WMMA_F32(c00, a0, b0); WMMA_F32(c01, a0, b1);
        WMMA_F32(c10, a1, b0); WMMA_F32(c11, a1, b1);
    }
    float* O = xw + (size_t)b * MM * FINN;
    for (int j = 0; j < 8; ++j) {
        const int row0 = ib * 32 + j + 8 * half, row1 = row0 + 16;
        O[row0 * FINN + bcol0] = c00[j];
        O[row0 * FINN + bcol1] = c01[j];
        O[row1 * FINN + bcol0] = c10[j];
        O[row1 * FINN + bcol1] = c11[j];
    }
}

// ============ Kernel 2: sq[row] = sum_f xw[row,f]^2 ============
__global__ void k_sq(const float* __restrict__ xw, float* __restrict__ sq) {
    const int row = blockIdx.x * blockDim.x + threadIdx.x;   // 0..B*M-1
    const float4* p = (const float4*)(xw + (size_t)row * FINN);
    float acc = 0.0f;
    for (int i = 0; i < FINN / 4; ++i) {
        float4 v = p[i];
        acc += v.x * v.x + v.y * v.y + v.z * v.z + v.w * v.w;
    }
    sq[row] = acc;
}

// ============ Kernel 3: G = xw @ xw^T, W = exp(-dist), 2x2 blocked ============
// blocks: 16 x 16 of 32x32 ; waves = B*16*16
__global__ void k_gramW(const float* __restrict__ xw, const float* __restrict__ sq,
                        float* __restrict__ Wout) {
    const int lane = threadIdx.x & 31;
    const int wave = threadIdx.x >> 5;
    const int t    = blockIdx.x * 4 + wave;
    const int b    = t >> 8;            // / (16*16)
    const int ib   = (t >> 4) & 15;
    const int jb   = t & 15;
    const int half = lane >> 4;
    const int r    = lane & 15;

    const float* XW = xw + (size_t)b * MM * FINN;
    const int arow0 = ib * 32 + r, arow1 = arow0 + 16;
    const int brow0 = jb * 32 + r, brow1 = brow0 + 16;   // B[k][col] = xw[col][k]
    v8f c00 = {}, c01 = {}, c10 = {}, c11 = {};
    for (int k = 0; k < FINN; k += 4) {
        const int kk = k + 2 * half;
        v2f a0, a1, b0, b1;
        a0[0] = XW[arow0 * FINN + kk];  a0[1] = XW[arow0 * FINN + kk + 1];
        a1[0] = XW[arow1 * FINN + kk];  a1[1] = XW[arow1 * FINN + kk + 1];
        b0[0] = XW[brow0 * FINN + kk];  b0[1] = XW[brow0 * FINN + kk + 1];
        b1[0] = XW[brow1 * FINN + kk];  b1[1] = XW[brow1 * FINN + kk + 1];
        WMMA_F32(c00, a0, b0); WMMA_F32(c01, a0, b1);
        WMMA_F32(c10, a1, b0); WMMA_F32(c11, a1, b1);
    }
    const float* SQ = sq + (size_t)b * MM;
    float* WB = Wout + (size_t)b * MM * MM;
    const int col0 = jb * 32 + r, col1 = col0 + 16;
    const float sqc0 = SQ[col0], sqc1 = SQ[col1];
    for (int j = 0; j < 8; ++j) {
        const int row0 = ib * 32 + j + 8 * half, row1 = row0 + 16;
        const float sr0 = SQ[row0], sr1 = SQ[row1];
        WB[(size_t)row0 * MM + col0] = expf(-sqrtf(fmaxf(sr0 + sqc0 - 2.0f * c00[j], 1e-12f)));
        WB[(size_t)row0 * MM + col1] = expf(-sqrtf(fmaxf(sr0 + sqc1 - 2.0f * c01[j], 1e-12f)));
        WB[(size_t)row1 * MM + col0] = expf(-sqrtf(fmaxf(sr1 + sqc0 - 2.0f * c10[j], 1e-12f)));
        WB[(size_t)row1 * MM + col1] = expf(-sqrtf(fmaxf(sr1 + sqc1 - 2.0f * c11[j], 1e-12f)));
    }
}

// ============ Kernel 4: dis[row] = rsqrt(rowsum(W) + 1e-7) ============
// one wave per row; float4 strided loads; deterministic tree reduction
__global__ void k_dis(const float* __restrict__ Wout, float* __restrict__ dis) {
    const int lane = threadIdx.x & 31;
    const int wave = threadIdx.x >> 5;
    const int row  = blockIdx.x * 4 + wave;     // global row = b*M + m
    const float4* Wr = (const float4*)(Wout + (size_t)row * MM);
    float s = 0.0f;
    for (int c = lane; c < MM / 4; c += 32) {
        float4 v = Wr[c];
        s += v.x + v.y + v.z + v.w;
    }
    for (int off = 16; off > 0; off >>= 1) s += __shfl_down(s, off, 32);
    if (lane == 0) dis[row] = rsqrtf(s + 1e-7f);
}

// ============ Kernel 5: L = I - dis_i*W*dis_j, + fused ||L||^2 partials ============
// float4 per thread; each block covers 1024 contiguous elements of one molecule
__global__ void k_L(const float* __restrict__ Wout, const float* __restrict__ dis,
                    float* __restrict__ Lout, float* __restrict__ partial) {
    __shared__ float red[256];
    const unsigned base = (blockIdx.x * 256u + threadIdx.x) * 4u;  // element index
    const unsigned b    = base / (MM * MM);
    const unsigned rem  = base % (MM * MM);
    const unsigned i    = rem / MM;
    const unsigned j0   = rem % MM;                // multiple of 4
    const float di = dis[b * MM + i];
    const float4 w  = *(const float4*)(Wout + base);
    const float4 dj = *(const float4*)(dis + b * MM + j0);
    float4 l;
    l.x = (i == j0     ? 1.0f : 0.0f) - di * w.x * dj.x;
    l.y = (i == j0 + 1 ? 1.0f : 0.0f) - di * w.y * dj.y;
    l.z = (i == j0 + 2 ? 1.0f : 0.0f) - di * w.z * dj.z;
    l.w = (i == j0 + 3 ? 1.0f : 0.0f) - di * w.w * dj.w;
    *(float4*)(Lout + base) = l;
    red[threadIdx.x] = l.x * l.x + l.y * l.y + l.z * l.z + l.w * l.w;
    __syncthreads();
    for (int s = 128; s > 0; s >>= 1) {
        if (threadIdx.x < (unsigned)s) red[threadIdx.x] += red[threadIdx.x + s];
        __syncthreads();
    }
    if (threadIdx.x == 0) partial[blockIdx.x] = red[0];
}

// ============ Kernel 6: factor[b] from 256 partials per molecule ============
__global__ void k_factor(const float* __restrict__ partial, float* __restrict__ factor) {
    __shared__ float red[256];
    const int b = blockIdx.x;
    red[threadIdx.x] = partial[b * 256 + threadIdx.x];
    __syncthreads();
    for (int s = 128; s > 0; s >>= 1) {
        if (threadIdx.x < s) red[threadIdx.x] += red[threadIdx.x + s];
        __syncthreads();
    }
    if (threadIdx.x == 0) {
        const float avg = sqrtf(red[0]) / (float)(MM * MM);
        factor[b] = fminf(1.0f / (avg + 1e-6f), 1.0f);
    }
}

// ============ Kernel 7: res_L = leaky(L * factor), in place, float4 ============
__global__ void k_resL(float* __restrict__ Lout, const float* __restrict__ factor) {
    const unsigned base = (blockIdx.x * 256u + threadIdx.x) * 4u;
    const unsigned b = base / (MM * MM);
    const float f = factor[b];
    float4 v = *(float4*)(Lout + base);
    v.x = leaky(v.x * f); v.y = leaky(v.y * f);
    v.z = leaky(v.z * f); v.w = leaky(v.w * f);
    *(float4*)(Lout + base) = v;
}

// ============ Kernel 8: x1 = (res_L + La) @ x, 2x2 blocked, K=512 ============
__global__ void k_x1(const float* __restrict__ resL, const float* __restrict__ La,
                     const float* __restrict__ x, float* __restrict__ x1) {
    const int lane = threadIdx.x & 31;
    const int wave = threadIdx.x >> 5;
    const int t    = blockIdx.x * 4 + wave;
    const int b    = t >> 6;            // / (16*4)
    const int ib   = (t >> 2) & 15;
    const int jb   = t & 3;
    const int half = lane >> 4;
    const int r    = lane & 15;

    const float* Lr  = resL + (size_t)b * MM * MM;
    const float* Lab = La   + (size_t)b * MM * MM;
    const float* X   = x    + (size_t)b * MM * FINN;
    const int arow0 = ib * 32 + r, arow1 = arow0 + 16;
    const int bcol0 = jb * 32 + r, bcol1 = bcol0 + 16;
    v8f c00 = {}, c01 = {}, c10 = {}, c11 = {};
    for (int k = 0; k < MM; k += 4) {
        const int kk = k + 2 * half;
        v2f a0, a1, b0, b1;
        a0[0] = Lr[arow0 * MM + kk]     + Lab[arow0 * MM + kk];
        a0[1] = Lr[arow0 * MM + kk + 1] + Lab[arow0 * MM + kk + 1];
        a1[0] = Lr[arow1 * MM + kk]     + Lab[arow1 * MM + kk];
        a1[1] = Lr[arow1 * MM + kk + 1] + Lab[arow1 * MM + kk + 1];
        b0[0] = X[kk * FINN + bcol0];   b0[1] = X[(kk + 1) * FINN + bcol0];
        b1[0] = X[kk * FINN + bcol1];   b1[1] = X[(kk + 1) * FINN + bcol1];
        WMMA_F32(c00, a0, b0); WMMA_F32(c01, a0, b1);
        WMMA_F32(c10, a1, b0); WMMA_F32(c11, a1, b1);
    }
    float* O = x1 + (size_t)b * MM * FINN;
    for (int j = 0; j < 8; ++j) {
        const int row0 = ib * 32 + j + 8 * half, row1 = row0 + 16;
        O[row0 * FINN + bcol0] = c00[j];
        O[row0 * FINN + bcol1] = c01[j];
        O[row1 * FINN + bcol0] = c10[j];
        O[row1 * FINN + bcol1] = c11[j];
    }
}

// ============ Kernel 9: out = leaky(xk @ weight + bias), 2x2 blocked, K=256 ============
// xk[m, 2f] = x[m,f] ; xk[m, 2f+1] = x1[m,f] ; kk always even -> a=[x, x1]
__global__ void k_out(const float* __restrict__ x, const float* __restrict__ x1,
                      const float* __restrict__ weight, const float* __restrict__ bias,
                      float* __restrict__ outp) {
    const int lane = threadIdx.x & 31;
    const int wave = threadIdx.x >> 5;
    const int t    = blockIdx.x * 4 + wave;
    const int b    = t >> 6;
    const int ib   = (t >> 2) & 15;
    const int ob   = t & 3;
    const int half = lane >> 4;
    const int r    = lane & 15;

    const float* X0 = x  + (size_t)b * MM * FINN;
    const float* X1 = x1 + (size_t)b * MM * FINN;
    const int arow0 = ib * 32 + r, arow1 = arow0 + 16;
    const int bcol0 = ob * 32 + r, bcol1 = bcol0 + 16;
    v8f c00 = {}, c01 = {}, c10 = {}, c11 = {};
    for (int k = 0; k < 2 * FINN; k += 4) {
        const int kk = k + 2 * half;       // even
        const int f  = kk >> 1;
        v2f a0, a1, b0, b1;
        a0[0] = X0[arow0 * FINN + f];   a0[1] = X1[arow0 * FINN + f];
        a1[0] = X0[arow1 * FINN + f];   a1[1] = X1[arow1 * FINN + f];
        b0[0] = weight[kk * OUTN + bcol0];  b0[1] = weight[(kk + 1) * OUTN + bcol0];
        b1[0] = weight[kk * OUTN + bcol1];  b1[1] = weight[(kk + 1) * OUTN + bcol1];
        WMMA_F32(c00, a0, b0); WMMA_F32(c01, a0, b1);
        WMMA_F32(c10, a1, b0); WMMA_F32(c11, a1, b1);
    }
    float* O = outp + (size_t)b * MM * OUTN;
    const float bv0 = bias[bcol0], bv1 = bias[bcol1];
    for (int j = 0; j < 8; ++j) {
        const int row0 = ib * 32 + j + 8 * half, row1 = row0 + 16;
        O[row0 * OUTN + bcol0] = leaky(c00[j] + bv0);
        O[row0 * OUTN + bcol1] = leaky(c01[j] + bv1);
        O[row1 * OUTN + bcol0] = leaky(c10[j] + bv0);
        O[row1 * OUTN + bcol1] = leaky(c11[j] + bv1);
    }
}

extern "C" void kernel_launch(void* const* d_in, const int* in_sizes, int n_in,
                              void* d_out, int out_size, void* d_ws, size_t ws_size,
                              hipStream_t stream) {
    (void)in_sizes; (void)n_in; (void)out_size; (void)ws_size;
    const float* x      = (const float*)d_in[0];
    const float* La     = (const float*)d_in[1];
    const float* weight = (const float*)d_in[2];
    const float* bias   = (const float*)d_in[3];
    const float* ML     = (const float*)d_in[4];

    // d_out = [out (B*M*OUT), res_L (B*M*M), W (B*M*M)] concatenated flat
    float* outp = (float*)d_out;
    float* Lout = outp + (size_t)BMOL * MM * OUTN;
    float* Wout = Lout + (size_t)BMOL * MM * MM;

    // workspace layout
    float* ws      = (float*)d_ws;
    float* xw      = ws;                                // B*M*FIN
    float* x1      = xw  + (size_t)BMOL * MM * FINN;    // B*M*FIN
    float* sq      = x1  + (size_t)BMOL * MM * FINN;    // B*M
    float* dis     = sq  + (size_t)BMOL * MM;           // B*M
    float* factor  = dis + (size_t)BMOL * MM;           // B
    float* partial = factor + BMOL;                     // B*M*M/1024 = 16384

    const int nLblk = BMOL * MM * MM / 1024;            // 16384 blocks, 1024 elems each

    k_xw    <<<BMOL * 16 * 4  / 4, 128, 0, stream>>>(x, ML, xw);
    k_sq    <<<BMOL * MM / 256,    256, 0, stream>>>(xw, sq);
    k_gramW <<<BMOL * 16 * 16 / 4, 128, 0, stream>>>(xw, sq, Wout);
    k_dis   <<<BMOL * MM / 4,      128, 0, stream>>>(Wout, dis);
    k_L     <<<nLblk,              256, 0, stream>>>(Wout, dis, Lout, partial);
    k_factor<<<BMOL,               256, 0, stream>>>(partial, factor);
    k_resL  <<<nLblk,              256, 0, stream>>>(Lout, factor);
    k_x1    <<<BMOL * 16 * 4 / 4,  128, 0, stream>>>(Lout, La, x, x1);
    k_out   <<<BMOL * 16 * 4 / 4,  128, 0, stream>>>(x, x1, weight, bias, outp);
}